// DeformableSelfAttention_71923522339271
// MI455X (gfx1250) — compile-verified
//
#include <hip/hip_runtime.h>
#include <hip/hip_bf16.h>
#include <cstdint>
#include <cstddef>

// ---------------------------------------------------------------------------
// DeformableSelfAttention, MI455X (gfx1250, wave32).
//   B=4, N=8192, C=1024, H=8, P=4.
// Key simplifications (exact, not approximations):
//   * final GEMM rows are constant => out[t][j] = s[t]*colsum(W_out)[j]+b_out[j]
//   * off[...,1] unused => only 32 even columns of W_off needed
// Memory floor: read x (128MB) + write out (128MB) ~= 11us @ 23.3 TB/s.
// Projection GEMM (32768x1024x64, fp32) done with V_WMMA_F32_16X16X4_F32.
// ---------------------------------------------------------------------------

typedef float v2f __attribute__((ext_vector_type(2)));
typedef float v8f __attribute__((ext_vector_type(8)));

#define C_DIM        1024
#define H_HEADS      8
#define P_PTS        4
#define N_PER_BATCH  8192
#define TOK_TOTAL    32768          // B*N
#define NCOLS        64             // 32 idx-logits | 32 attn-logits
#define KSTEPS       256            // C/4
#define TOK_PER_BLK  64
#define ROW_TILES    4              // TOK_PER_BLK / 16
#define LOG_STRIDE   66             // 64 + pad, LDS bank-conflict avoidance

// ---------------------------------------------------------------- prep: wsum
__global__ __launch_bounds__(128) void prep_wsum(const float* __restrict__ W_out,
                                                 float* __restrict__ wsum) {
    int j = blockIdx.x * 128 + threadIdx.x;         // 0..1023, coalesced
    float a = 0.f;
    for (int i = 0; i < C_DIM; ++i) a += W_out[(size_t)i * C_DIM + j];
    wsum[j] = a;
}

// ------------------------------------- prep: pack weights in B-fragment order
// B-frag (16x16x4 f32): lane l holds Wcat[k0 + 2*(l>>4) + {0,1}][16*g + (l&15)]
// Packed as float2 at gid = t*128 + g*32 + lane  -> perfectly coalesced loads.
__global__ __launch_bounds__(256) void prep_pack(const float* __restrict__ W_off,
                                                 const float* __restrict__ W_attn,
                                                 const float* __restrict__ b_off,
                                                 const float* __restrict__ b_attn,
                                                 float* __restrict__ Bpack,
                                                 float* __restrict__ bcat) {
    int gid  = blockIdx.x * 256 + threadIdx.x;      // 0..32767
    int t    = gid >> 7;                            // k-step
    int rem  = gid & 127;
    int g    = rem >> 5;                            // wave / col-group
    int lane = rem & 31;
    int k    = t * 4 + ((lane >> 4) << 1);
    int c    = g * 16 + (lane & 15);                // logical col 0..63
    float v0, v1;
    if (c < 32) {                                   // idx logits: even W_off cols
        v0 = W_off[(size_t)k * 64 + 2 * c];
        v1 = W_off[(size_t)(k + 1) * 64 + 2 * c];
    } else {
        int cc = c - 32;
        v0 = W_attn[(size_t)k * 32 + cc];
        v1 = W_attn[(size_t)(k + 1) * 32 + cc];
    }
    Bpack[2 * gid]     = v0;
    Bpack[2 * gid + 1] = v1;
    if (gid < 64) bcat[gid] = (gid < 32) ? b_off[2 * gid] : b_attn[gid - 32];
}

// ------------------------------------------------------ fused projection + s
__global__ __launch_bounds__(128) void fused_proj(const float* __restrict__ x,
                                                  const float* __restrict__ Bpack,
                                                  const float* __restrict__ bcat,
                                                  float* __restrict__ s_out) {
    __shared__ float x0s[C_DIM];                    // x[batch][0][:]
    __shared__ float bcs[NCOLS];
    __shared__ float logits[TOK_PER_BLK * LOG_STRIDE];

    const int tid  = threadIdx.x;                   // 0..127 (4 waves)
    const int wave = tid >> 5;
    const int lane = tid & 31;
    const int lrow = lane & 15;
    const int khalf = (lane >> 4) << 1;             // 0 or 2

    const size_t tokbase = (size_t)blockIdx.x * TOK_PER_BLK;
    const int batch = (int)(tokbase >> 13);         // /8192; 64 | 8192 so no straddle

    // stage x0 (4KB) and biases
    const float* xb0 = x + (size_t)batch * N_PER_BATCH * C_DIM;
    for (int i = tid; i < C_DIM; i += 128) x0s[i] = xb0[i];
    if (tid < NCOLS) bcs[tid] = bcat[tid];

    v8f acc[ROW_TILES];
    #pragma unroll
    for (int r = 0; r < ROW_TILES; ++r) acc[r] = (v8f)(0.f);

    const float* xA = x + tokbase * C_DIM;
    for (int t = 0; t < KSTEPS; ++t) {
        const int k0 = t * 4;
        // B fragment: pre-packed, coalesced 8B/lane
        v2f bfrag = *(const v2f*)(Bpack + (size_t)2 * ((size_t)t * 128 + wave * 32 + lane));
        #pragma unroll
        for (int r = 0; r < ROW_TILES; ++r) {
            // A fragment: lane l -> x[tok = base + r*16 + (l&15)][k0 + khalf .. +1]
            v2f afrag = *(const v2f*)(xA + (size_t)(r * 16 + lrow) * C_DIM + k0 + khalf);
            acc[r] = __builtin_amdgcn_wmma_f32_16x16x4_f32(
                false, afrag, false, bfrag, (short)0, acc[r], false, false);
        }
    }

    // spill C/D fragments: VGPR i -> M = i (lanes 0-15) or i+8 (lanes 16-31), N = lane&15
    const int colbase = wave * 16;
    #pragma unroll
    for (int r = 0; r < ROW_TILES; ++r) {
        #pragma unroll
        for (int i = 0; i < 8; ++i) {
            int row = r * 16 + ((lane < 16) ? i : (i + 8));
            logits[row * LOG_STRIDE + colbase + lrow] = acc[r][i];
        }
    }
    __syncthreads();

    // epilogue: one token per thread (tid < 64)
    if (tid < TOK_PER_BLK) {
        const float* lg = &logits[tid * LOG_STRIDE];
        float s = 0.f;
        #pragma unroll
        for (int h = 0; h < H_HEADS; ++h) {
            float a0 = lg[32 + 4 * h + 0] + bcs[32 + 4 * h + 0];
            float a1 = lg[32 + 4 * h + 1] + bcs[32 + 4 * h + 1];
            float a2 = lg[32 + 4 * h + 2] + bcs[32 + 4 * h + 2];
            float a3 = lg[32 + 4 * h + 3] + bcs[32 + 4 * h + 3];
            float m  = fmaxf(fmaxf(a0, a1), fmaxf(a2, a3));
            float e0 = __expf(a0 - m), e1 = __expf(a1 - m);
            float e2 = __expf(a2 - m), e3 = __expf(a3 - m);
            float den = e0 + e1 + e2 + e3;
            float num = 0.f;
            #pragma unroll
            for (int p = 0; p < P_PTS; ++p) {
                float f = lg[4 * h + p] + bcs[4 * h + p];
                int idx = (int)f;                       // trunc toward zero (matches astype)
                if (idx < 0) idx += C_DIM;              // wrap
                idx = (idx < 0) ? 0 : ((idx > C_DIM - 1) ? C_DIM - 1 : idx);
                float e = (p == 0) ? e0 : (p == 1) ? e1 : (p == 2) ? e2 : e3;
                num += x0s[idx] * e;
            }
            s += num / den;
        }
        s_out[tokbase + tid] = s;
    }
}

// --------------------------------------------- rank-1 broadcast output stream
__global__ __launch_bounds__(256) void bcast_out(const float* __restrict__ s,
                                                 const float* __restrict__ wsum,
                                                 const float* __restrict__ b_out,
                                                 float* __restrict__ out) {
    const size_t t = blockIdx.x;                    // token
    const int i = threadIdx.x;                      // 256 threads * float4 = 1024
    const float sv = s[t];
    const float4 w  = ((const float4*)wsum)[i];
    const float4 bb = ((const float4*)b_out)[i];
    float4 o;
    o.x = fmaf(sv, w.x, bb.x);
    o.y = fmaf(sv, w.y, bb.y);
    o.z = fmaf(sv, w.z, bb.z);
    o.w = fmaf(sv, w.w, bb.w);
    ((float4*)(out + t * C_DIM))[i] = o;
}

// ---------------------------------------------------------------------------
extern "C" void kernel_launch(void* const* d_in, const int* in_sizes, int n_in,
                              void* d_out, int out_size, void* d_ws, size_t ws_size,
                              hipStream_t stream) {
    (void)in_sizes; (void)n_in; (void)out_size; (void)ws_size;
    const float* x      = (const float*)d_in[0];
    const float* W_off  = (const float*)d_in[1];
    const float* b_off  = (const float*)d_in[2];
    const float* W_attn = (const float*)d_in[3];
    const float* b_attn = (const float*)d_in[4];
    const float* W_out  = (const float*)d_in[5];
    const float* b_out  = (const float*)d_in[6];
    float* out = (float*)d_out;

    char* ws = (char*)d_ws;
    float* s_ws  = (float*)(ws);               // 32768 floats  (128 KB)
    float* wsum  = (float*)(ws + 131072);      // 1024 floats   (4 KB)
    float* bcat  = (float*)(ws + 135168);      // 64 floats
    float* Bpack = (float*)(ws + 135424);      // 65536 floats  (256 KB)

    hipLaunchKernelGGL(prep_wsum, dim3(8),   dim3(128), 0, stream, W_out, wsum);
    hipLaunchKernelGGL(prep_pack, dim3(128), dim3(256), 0, stream,
                       W_off, W_attn, b_off, b_attn, Bpack, bcat);
    hipLaunchKernelGGL(fused_proj, dim3(TOK_TOTAL / TOK_PER_BLK), dim3(128), 0, stream,
                       x, Bpack, bcat, s_ws);
    hipLaunchKernelGGL(bcast_out, dim3(TOK_TOTAL), dim3(256), 0, stream,
                       s_ws, wsum, b_out, out);
}